// CTCLayer_53498112639193
// MI455X (gfx1250) — compile-verified
//
#include <hip/hip_runtime.h>
#include <hip/hip_bf16.h>
#include <stdint.h>

// CTC forward loss on gfx1250 (MI455X).
// B=1024, T=384, V=60, L=45, S=2L+1=91, BLANK=59.
// Shape: one wave32 per batch element, ZERO workgroup barriers.
//  - alpha[0..90] in 3 VGPRs/lane (91 = 32+32+27); banded neighbors via
//    in-wave rotate shuffles (ds_bpermute).
//  - per-row logsumexp = wave32 butterfly (__shfl_xor).
//  - logits rows streamed into an 8-slot LDS ring with
//    global_load_async_to_lds_b128 (one async instruction per row, ASYNCcnt
//    in-order completion): 7 rows in flight, fenced with s_wait_asynccnt 7,
//    so ~7 steps of compute cover HBM latency on the serial-in-T chain.

#define CTC_B     1024
#define CTC_T     384
#define CTC_V     60
#define CTC_L     45
#define CTC_S     91        // 2*L+1
#define CTC_BLANK 59
#define CTC_NEG   (-1.0e30f)
#define NBUF      8         // LDS ring slots (power of two)

__device__ __forceinline__ float lse3(float a, float b, float c) {
  float m = fmaxf(fmaxf(a, b), c);
  return m + __logf(__expf(a - m) + __expf(b - m) + __expf(c - m));
}

__global__ __launch_bounds__(32) void ctc_forward_kernel(
    const int*   __restrict__ labels,   // [B, L]
    const float* __restrict__ logits,   // [B, T, V]
    float*       __restrict__ nll_out)  // [B]
{
  const int b    = blockIdx.x;
  const int lane = threadIdx.x;          // wave32: lane == tid

  __shared__ float ring[NBUF][64];       // 8-slot ring of logits rows (60 used)

  const float* rowbase = logits + (size_t)b * CTC_T * CTC_V;
  const int    bL      = b * CTC_L;

  // ---- per-lane static CTC state for positions s = lane, lane+32, lane+64 ----
  auto extAt = [&](int s) -> int {
    if (s >= CTC_S) return CTC_BLANK;
    return (s & 1) ? labels[bL + (s >> 1)] : CTC_BLANK;
  };
  const int s2ok = (lane < CTC_S - 64);  // lane < 27
  const int e0 = extAt(lane);
  const int e1 = extAt(lane + 32);
  const int e2 = s2ok ? extAt(lane + 64) : 0;
  auto skipAt = [&](int s, int es) -> int {
    return (s >= 2) && (es != CTC_BLANK) && (es != extAt(s - 2));
  };
  const int sk0 = skipAt(lane, e0);
  const int sk1 = skipAt(lane + 32, e1);
  const int sk2 = s2ok ? skipAt(lane + 64, e2) : 0;

  // label_len via two wave ballots (45 labels > 32 lanes).
  int nz_lo = (lane < 32) && (labels[bL + lane] != CTC_BLANK);
  int nz_hi = (lane < CTC_L - 32) && (labels[bL + 32 + lane] != CTC_BLANK);
  const int llen = __popcll(__ballot(nz_lo)) + __popcll(__ballot(nz_hi));

  // One async instruction stages one 60-float row (15 lanes x B128).
  auto issueRow = [&](int t) {
    if (lane < 15) {
      uint32_t lds_addr = (uint32_t)(uintptr_t)(&ring[t & (NBUF - 1)][lane * 4]);
      uint32_t goff     = (uint32_t)(t * CTC_V * 4 + lane * 16);
      asm volatile("global_load_async_to_lds_b128 %0, %1, %2"
                   :: "v"(lds_addr), "v"(goff), "s"(rowbase) : "memory");
    }
  };

  const unsigned lm1 = (lane + 31u) & 31u;   // rotate-by-1 source lane
  const unsigned lm2 = (lane + 30u) & 31u;   // rotate-by-2 source lane

  float a0 = CTC_NEG, a1 = CTC_NEG, a2 = CTC_NEG;

  // One CTC time step: LSE of row, emission gather, banded logaddexp update.
  auto step = [&](int t) {
    const float* rw = ring[t & (NBUF - 1)];
    float x0 = rw[lane];
    float x1 = (lane + 32 < CTC_V) ? rw[lane + 32] : CTC_NEG;
    float mx = fmaxf(x0, x1);
    #pragma unroll
    for (int o = 16; o > 0; o >>= 1) mx = fmaxf(mx, __shfl_xor(mx, o, 32));
    float sm = __expf(x0 - mx) + ((lane + 32 < CTC_V) ? __expf(x1 - mx) : 0.0f);
    #pragma unroll
    for (int o = 16; o > 0; o >>= 1) sm += __shfl_xor(sm, o, 32);
    const float d = mx + __logf(sm);

    const float em0 = rw[e0] - d;
    const float em1 = rw[e1] - d;
    const float em2 = rw[e2] - d;

    if (t == 0) {
      a0 = (lane <= 1) ? em0 : CTC_NEG;        // only s=0,1 reachable at t=0
    } else {
      float r0 = __shfl(a0, lm1, 32), r1 = __shfl(a1, lm1, 32), r2 = __shfl(a2, lm1, 32);
      float q0 = __shfl(a0, lm2, 32), q1 = __shfl(a1, lm2, 32), q2 = __shfl(a2, lm2, 32);
      float p1_0 = (lane == 0) ? CTC_NEG : r0;
      float p1_1 = (lane == 0) ? r0 : r1;      // lane0: pos 31 = reg0@31
      float p1_2 = (lane == 0) ? r1 : r2;      // lane0: pos 63 = reg1@31
      float p2_0 = (lane < 2) ? CTC_NEG : q0;
      float p2_1 = (lane < 2) ? q0 : q1;       // lanes 0,1: pos 30,31 = reg0@30,31
      float p2_2 = (lane < 2) ? q1 : q2;       // lanes 0,1: pos 62,63 = reg1@30,31
      p2_0 = sk0 ? p2_0 : CTC_NEG;
      p2_1 = sk1 ? p2_1 : CTC_NEG;
      p2_2 = sk2 ? p2_2 : CTC_NEG;

      a0 = lse3(a0, p1_0, p2_0) + em0;
      a1 = lse3(a1, p1_1, p2_1) + em1;
      float na2 = lse3(a2, p1_2, p2_2) + em2;
      a2 = s2ok ? na2 : CTC_NEG;
    }
  };

  // Prologue: fill the pipeline with rows 0..NBUF-2 (7 in flight).
  #pragma unroll
  for (int t = 0; t < NBUF - 1; ++t) issueRow(t);

  // Main loop: keep 8 outstanding; cnt<=7 + in-order completion => row t resident.
  int t = 0;
  for (; t < CTC_T - (NBUF - 1); ++t) {
    issueRow(t + NBUF - 1);
    asm volatile("s_wait_asynccnt 7" ::: "memory");
    step(t);
  }
  // Tail: everything issued; drain once, then run wait-free.
  asm volatile("s_wait_asynccnt 0" ::: "memory");
  for (; t < CTC_T; ++t) step(t);

  // Gather alpha at end = 2*label_len and end-1 (uniform across the wave).
  const int end = 2 * llen;
  auto getpos = [&](int p) -> float {
    float v0 = __shfl(a0, p & 31, 32);
    float v1 = __shfl(a1, p & 31, 32);
    float v2 = __shfl(a2, p & 31, 32);
    int r = p >> 5;
    return (r == 0) ? v0 : ((r == 1) ? v1 : v2);
  };
  float ab = getpos(end);
  float al = (end > 0) ? getpos(end - 1) : CTC_NEG;
  if (lane == 0) {
    float m   = fmaxf(ab, al);
    float nll = -(m + __logf(__expf(ab - m) + __expf(al - m)));
    nll_out[b] = nll;
  }
}

// Deterministic single-block mean over B=1024 per-batch losses.
__global__ __launch_bounds__(256) void ctc_mean_kernel(
    const float* __restrict__ nll, float* __restrict__ out)
{
  __shared__ float sh[256];
  const int tid = threadIdx.x;
  float s = 0.0f;
  for (int i = tid; i < CTC_B; i += 256) s += nll[i];
  sh[tid] = s;
  __syncthreads();
  #pragma unroll
  for (int o = 128; o > 0; o >>= 1) {
    if (tid < o) sh[tid] += sh[tid + o];
    __syncthreads();
  }
  if (tid == 0) out[0] = sh[0] * (1.0f / (float)CTC_B);
}

extern "C" void kernel_launch(void* const* d_in, const int* in_sizes, int n_in,
                              void* d_out, int out_size, void* d_ws, size_t ws_size,
                              hipStream_t stream) {
  (void)in_sizes; (void)n_in; (void)out_size; (void)ws_size;
  const int*   labels = (const int*)d_in[0];    // [B, L] int32
  const float* logits = (const float*)d_in[1];  // [B, T, V] f32
  float* nll = (float*)d_ws;                    // B floats of scratch

  ctc_forward_kernel<<<CTC_B, 32, 0, stream>>>(labels, logits, nll);
  ctc_mean_kernel<<<1, 256, 0, stream>>>(nll, (float*)d_out);
}